// EdgeNet_7456063226140
// MI455X (gfx1250) — compile-verified
//
#include <hip/hip_runtime.h>
#include <math.h>

typedef float v2f __attribute__((ext_vector_type(2)));
typedef float v4f __attribute__((ext_vector_type(4)));
typedef float v8f __attribute__((ext_vector_type(8)));

#define BN_EPS 1e-5f

// D = A(16x4 f32) * B(4x16 f32) + C(16x16 f32)   -- full fp32 WMMA on CDNA5
__device__ __forceinline__ v8f wmma4(v2f a, v2f b, v8f c) {
  return __builtin_amdgcn_wmma_f32_16x16x4_f32(
      /*neg_a=*/false, a, /*neg_b=*/false, b,
      /*c_mod=*/(short)0, c, /*reuse_a=*/false, /*reuse_b=*/false);
}

// Same-wave LDS store -> cross-lane load ordering (DS ops are in-order per
// wave in HW; this also stops the compiler from reordering across it).
__device__ __forceinline__ void lds_fence() {
  asm volatile("s_wait_dscnt 0" ::: "memory");
}

// ---------------- BatchNorm ----------------
__global__ void __launch_bounds__(256) bn_stats_kernel(const float* __restrict__ x,
                                                       float* __restrict__ stats, int N) {
  __shared__ float red[256][8];
  float s0 = 0, s1 = 0, s2 = 0, s3 = 0, q0 = 0, q1 = 0, q2 = 0, q3 = 0;
  for (int i = (int)threadIdx.x; i < N; i += 256) {
    v4f v = *(const v4f*)(x + (size_t)i * 4);
    s0 += v.x; s1 += v.y; s2 += v.z; s3 += v.w;
    q0 += v.x * v.x; q1 += v.y * v.y; q2 += v.z * v.z; q3 += v.w * v.w;
  }
  red[threadIdx.x][0] = s0; red[threadIdx.x][1] = s1;
  red[threadIdx.x][2] = s2; red[threadIdx.x][3] = s3;
  red[threadIdx.x][4] = q0; red[threadIdx.x][5] = q1;
  red[threadIdx.x][6] = q2; red[threadIdx.x][7] = q3;
  __syncthreads();
  for (int off = 128; off > 0; off >>= 1) {
    if ((int)threadIdx.x < off) {
#pragma unroll
      for (int c = 0; c < 8; ++c) red[threadIdx.x][c] += red[threadIdx.x + off][c];
    }
    __syncthreads();
  }
  if (threadIdx.x < 4) {
    float mu  = red[0][threadIdx.x] / (float)N;
    float var = red[0][4 + threadIdx.x] / (float)N - mu * mu;
    stats[threadIdx.x]     = mu;
    stats[4 + threadIdx.x] = rsqrtf(var + BN_EPS);
  }
}

__global__ void bn_apply_kernel(const float* __restrict__ x, const float* __restrict__ stats,
                                const float* __restrict__ gamma, const float* __restrict__ beta,
                                float* __restrict__ xn, int total) {
  int i = blockIdx.x * blockDim.x + threadIdx.x;
  if (i < total) {
    int c = i & 3;
    xn[i] = (x[i] - stats[c]) * stats[4 + c] * gamma[c] + beta[c];
  }
}

// ---------------- degree / divide ----------------
__global__ void degree_kernel(const int* __restrict__ dst, float* __restrict__ cnt, int E) {
  int i = blockIdx.x * blockDim.x + threadIdx.x;
  int stride = gridDim.x * blockDim.x;
  for (; i < E; i += stride) atomicAdd(&cnt[dst[i]], 1.0f);
}

__global__ void divide_kernel(float* __restrict__ num, const float* __restrict__ cnt,
                              int N, int C) {
  int i = blockIdx.x * blockDim.x + threadIdx.x;
  if (i < N * C) num[i] /= fmaxf(cnt[i / C], 1.0f);
}

__global__ void divide_out_kernel(const float* __restrict__ acc, const float* __restrict__ cnt,
                                  float* __restrict__ out, int N, int C) {
  int i = blockIdx.x * blockDim.x + threadIdx.x;
  if (i < N * C) out[i] = acc[i] / fmaxf(cnt[i / C], 1.0f);
}

// ---------------- EdgeConv via WMMA ----------------
// One wave processes a tile of 16 edges. MLP: [KIN -> 32 -> 32 -> KOUT].
// A-layout (32-bit 16x4): lane m holds row M=m (m=lane&15); half=lane>>4
// selects K pair {2*half, 2*half+1}. B-layout mirrors: VGPR j holds row
// K = j + 2*half, cols across lanes. C-layout: VGPR v -> row v + 8*half.
template <int KIN, int KOUT, bool FRELU>
__global__ void __launch_bounds__(256) edgeconv_wmma(
    const float* __restrict__ feat,  // [N][KIN/2]
    const int* __restrict__ src, const int* __restrict__ dst,
    const float* __restrict__ W1, const float* __restrict__ b1,
    const float* __restrict__ W2, const float* __restrict__ b2,
    const float* __restrict__ W3, const float* __restrict__ b3,
    float* __restrict__ accum,  // [N][KOUT]
    int nEdges) {
  constexpr int D   = KIN / 2;   // per-node feature dim
  constexpr int KT1 = KIN / 4;   // k-tiles in layer 1
  __shared__ float lds[8][16][32];
  __shared__ int   dstLds[8][16];
  const int lane = threadIdx.x & 31;
  const int wib  = threadIdx.x >> 5;
  float (*act)[32] = lds[wib];
  int* dstIdx      = dstLds[wib];
  const int m    = lane & 15;
  const int half = lane >> 4;

  auto loadB = [&](const float* W, int ld, int kbase, int nbase, int nmax) -> v2f {
    int n = nbase + m;
    if (n >= nmax) n = nmax - 1;  // clamp: garbage cols never read back
    int k0 = kbase + 2 * half;
    v2f r;
    r.x = W[k0 * ld + n];
    r.y = W[(k0 + 1) * ld + n];
    return r;
  };

  // Preload all weight B-tiles into registers (loop-invariant).
  v2f B1[KT1][2], B2[8][2], B3[8];
#pragma unroll
  for (int kk = 0; kk < KT1; ++kk) {
    B1[kk][0] = loadB(W1, 32, 4 * kk, 0, 32);
    B1[kk][1] = loadB(W1, 32, 4 * kk, 16, 32);
  }
#pragma unroll
  for (int kk = 0; kk < 8; ++kk) {
    B2[kk][0] = loadB(W2, 32, 4 * kk, 0, 32);
    B2[kk][1] = loadB(W2, 32, 4 * kk, 16, 32);
  }
#pragma unroll
  for (int kk = 0; kk < 8; ++kk) B3[kk] = loadB(W3, KOUT, 4 * kk, 0, KOUT);
  const float bias1[2] = {b1[m], b1[16 + m]};
  const float bias2[2] = {b2[m], b2[16 + m]};
  const float bias3 = (m < KOUT) ? b3[m] : 0.0f;

  const int nTiles = (nEdges + 15) >> 4;
  const int wave   = (int)((blockIdx.x * blockDim.x + threadIdx.x) >> 5);
  const int nWaves = (int)((gridDim.x * blockDim.x) >> 5);

  for (int tile = wave; tile < nTiles; tile += nWaves) {
    const int ebase = tile << 4;
    // ---- gather 16 edges into LDS staging: [xi, xj - xi] ----
    if (half == 0) {
      const int e  = ebase + m;
      const bool ok = e < nEdges;
      int si = 0, di = 0;
      if (ok) { si = src[e]; di = dst[e]; }
      dstIdx[m] = ok ? di : -1;
      if constexpr (D == 4) {
        v4f xi = {}, xj = {};
        if (ok) {
          xi = *(const v4f*)(feat + (size_t)di * 4);
          xj = *(const v4f*)(feat + (size_t)si * 4);
        }
        *(v4f*)&act[m][0] = xi;       // row base: 128B aligned
        *(v4f*)&act[m][4] = xj - xi;  // 16B aligned
      } else {
        v2f xi = {}, xj = {};
        if (ok) {
          xi = *(const v2f*)(feat + (size_t)di * 2);
          xj = *(const v2f*)(feat + (size_t)si * 2);
        }
        *(v2f*)&act[m][0] = xi;
        *(v2f*)&act[m][2] = xj - xi;
      }
    }
    lds_fence();

    // ---- layer 1: [16,KIN] x [KIN,32] ----
    v8f c0 = {}, c1 = {};
#pragma unroll
    for (int kk = 0; kk < KT1; ++kk) {
      const v2f a = *(const v2f*)&act[m][4 * kk + 2 * half];  // 8B aligned
      c0 = wmma4(a, B1[kk][0], c0);
      c1 = wmma4(a, B1[kk][1], c1);
    }
    lds_fence();
#pragma unroll
    for (int v = 0; v < 8; ++v) {
      const int row = v + 8 * half;
      float t0 = c0[v] + bias1[0];
      float t1 = c1[v] + bias1[1];
      act[row][m]      = t0 > 0.f ? t0 : 0.f;
      act[row][16 + m] = t1 > 0.f ? t1 : 0.f;
    }
    lds_fence();

    // ---- layer 2: [16,32] x [32,32] ----
    v8f d0v = {}, d1v = {};
#pragma unroll
    for (int kk = 0; kk < 8; ++kk) {
      const v2f a = *(const v2f*)&act[m][4 * kk + 2 * half];
      d0v = wmma4(a, B2[kk][0], d0v);
      d1v = wmma4(a, B2[kk][1], d1v);
    }
    lds_fence();
#pragma unroll
    for (int v = 0; v < 8; ++v) {
      const int row = v + 8 * half;
      float t0 = d0v[v] + bias2[0];
      float t1 = d1v[v] + bias2[1];
      act[row][m]      = t0 > 0.f ? t0 : 0.f;
      act[row][16 + m] = t1 > 0.f ? t1 : 0.f;
    }
    lds_fence();

    // ---- layer 3: [16,32] x [32,KOUT] ----
    v8f o = {};
#pragma unroll
    for (int kk = 0; kk < 8; ++kk) {
      const v2f a = *(const v2f*)&act[m][4 * kk + 2 * half];
      o = wmma4(a, B3[kk], o);
    }
    // ---- scatter-add to destination nodes (mean numerator) ----
    if (m < KOUT) {
#pragma unroll
      for (int v = 0; v < 8; ++v) {
        const int row = v + 8 * half;
        const int d   = dstIdx[row];
        if (d >= 0) {
          float t = o[v] + bias3;
          if (FRELU) t = t > 0.f ? t : 0.f;
          atomicAdd(&accum[(size_t)d * KOUT + m], t);
        }
      }
    }
    lds_fence();
  }
}

extern "C" void kernel_launch(void* const* d_in, const int* in_sizes, int n_in,
                              void* d_out, int out_size, void* d_ws, size_t ws_size,
                              hipStream_t stream) {
  (void)n_in; (void)out_size; (void)ws_size;
  const float* x     = (const float*)d_in[0];
  const int*   edge  = (const int*)d_in[1];
  const float* gamma = (const float*)d_in[2];
  const float* beta  = (const float*)d_in[3];
  const float* eW1 = (const float*)d_in[4];  const float* eb1 = (const float*)d_in[5];
  const float* eW2 = (const float*)d_in[6];  const float* eb2 = (const float*)d_in[7];
  const float* eW3 = (const float*)d_in[8];  const float* eb3 = (const float*)d_in[9];
  const float* dW1 = (const float*)d_in[10]; const float* db1 = (const float*)d_in[11];
  const float* dW2 = (const float*)d_in[12]; const float* db2 = (const float*)d_in[13];
  const float* dW3 = (const float*)d_in[14]; const float* db3 = (const float*)d_in[15];

  const int N = in_sizes[0] / 4;
  const int E = in_sizes[1] / 2;
  const int* src = edge;      // edge_index[0]
  const int* dst = edge + E;  // edge_index[1]

  float* ws    = (float*)d_ws;
  float* xn    = ws;                   // N*4 normalized features
  float* h     = xn + (size_t)N * 4;   // N*2 encoder accum -> h
  float* cnt   = h + (size_t)N * 2;    // N   degrees
  float* acc   = cnt + (size_t)N;      // N*4 decoder accum
  float* stats = acc + (size_t)N * 4;  // 8   (mu[4], rstd[4])

  hipMemsetAsync(h, 0, sizeof(float) * (size_t)N * 2, stream);
  hipMemsetAsync(cnt, 0, sizeof(float) * (size_t)N, stream);
  hipMemsetAsync(acc, 0, sizeof(float) * (size_t)N * 4, stream);

  bn_stats_kernel<<<1, 256, 0, stream>>>(x, stats, N);
  {
    int total = N * 4;
    bn_apply_kernel<<<(total + 255) / 256, 256, 0, stream>>>(x, stats, gamma, beta, xn, total);
  }
  degree_kernel<<<2048, 256, 0, stream>>>(dst, cnt, E);

  const int convBlocks = 1024;  // 8192 waves, ~24 tiles/wave: amortizes weight preload
  edgeconv_wmma<8, 2, true><<<convBlocks, 256, 0, stream>>>(
      xn, src, dst, eW1, eb1, eW2, eb2, eW3, eb3, h, E);
  {
    int total = N * 2;
    divide_kernel<<<(total + 255) / 256, 256, 0, stream>>>(h, cnt, N, 2);
  }
  edgeconv_wmma<4, 4, false><<<convBlocks, 256, 0, stream>>>(
      h, src, dst, dW1, db1, dW2, db2, dW3, db3, acc, E);
  {
    int total = N * 4;
    divide_out_kernel<<<(total + 255) / 256, 256, 0, stream>>>(acc, cnt, (float*)d_out, N, 4);
  }
}